// CustomLoss_2181843386513
// MI455X (gfx1250) — compile-verified
//
#include <hip/hip_runtime.h>
#include <math.h>
#include <stdint.h>

// ---------------------------------------------------------------------------
// Problem constants (from reference): B=16384 rows, F=4096 features, fp32.
// Phase 1 is HBM-bandwidth bound (256 MB of train_data, ~11us at 23.3 TB/s).
// ---------------------------------------------------------------------------

#define EPS 1e-6f

typedef __attribute__((ext_vector_type(2))) float v2f;
typedef __attribute__((ext_vector_type(8))) float v8f;

#define ROWS_PER_WG 64
#define TILE_COLS   64
#define LPAD        4
#define LSTRIDE     (TILE_COLS + LPAD)   // 68 floats per LDS tile row

// ----- CDNA5 async global->LDS copy helpers (ASYNCcnt tracked) -------------
__device__ __forceinline__ uint32_t lds_off_of(const void* p) {
    // Flat pointers into the LDS aperture carry the LDS byte offset in the
    // low 32 bits (CDNA5 ISA 10.2 aperture mapping).
    return (uint32_t)(uintptr_t)p;
}

__device__ __forceinline__ void async_load_b128(uint32_t lds_off, const void* gaddr) {
    asm volatile("global_load_async_to_lds_b128 %0, %1, off"
                 :: "v"(lds_off), "v"((unsigned long long)(uintptr_t)gaddr)
                 : "memory");
}

__device__ __forceinline__ void async_wait_all() {
    asm volatile("s_wait_asynccnt 0" ::: "memory");
}

// ---------------------------------------------------------------------------
// Kernel 1: per-row diag[b] = exp(-sqrt( sum_f (test[f]-train[b][f]+eps)^2 ))
// 128 threads = 4 waves; wave w owns rows [wg*64 + 16w, +16).
// WMMA f32 16x16x4 with all-ones B performs the K reduction + accumulate.
// ---------------------------------------------------------------------------
__global__ __launch_bounds__(128) void dist_diag_kernel(
    const float* __restrict__ train, const float* __restrict__ test,
    float* __restrict__ diag, int F) {

    __shared__ float ltest[4096];                       // full test vector
    __shared__ float ltile[2][ROWS_PER_WG * LSTRIDE];   // double-buffered tile

    const int tid  = threadIdx.x;
    const int wave = tid >> 5;
    const int lane = tid & 31;
    const int row0 = blockIdx.x * ROWS_PER_WG;
    const int nTiles = F / TILE_COLS;

    // --- stage the (reused-by-all) test vector asynchronously ---
    #pragma unroll
    for (int s = 0; s < 8; ++s) {                       // 1024 b128 segs / 128 thr
        int seg = s * 128 + tid;
        async_load_b128(lds_off_of(&ltest[seg * 4]), test + seg * 4);
    }
    // --- stage tile 0 ---
    {
        #pragma unroll
        for (int s = 0; s < 8; ++s) {                   // 64 rows x 16 segs
            int seg = s * 128 + tid;
            int r   = seg >> 4;
            int ci  = (seg & 15) * 4;
            async_load_b128(lds_off_of(&ltile[0][r * LSTRIDE + ci]),
                            train + (size_t)(row0 + r) * F + ci);
        }
    }
    async_wait_all();
    __syncthreads();

    // A-matrix lane mapping (16x4 f32): lanes 0-15 -> K={0,1}, lanes 16-31 -> K={2,3}
    const int m       = lane & 15;
    const int koff    = (lane >> 4) * 2;
    const int tileRow = wave * 16 + m;

    const v2f ones = {1.0f, 1.0f};
    v8f c = {};                                          // 16x16 f32 accumulator

    for (int ti = 0; ti < nTiles; ++ti) {
        const int bb = ti & 1;
        // prefetch next tile into the other buffer (overlaps with compute)
        if (ti + 1 < nTiles) {
            #pragma unroll
            for (int s = 0; s < 8; ++s) {
                int seg = s * 128 + tid;
                int r   = seg >> 4;
                int ci  = (seg & 15) * 4;
                async_load_b128(lds_off_of(&ltile[bb ^ 1][r * LSTRIDE + ci]),
                                train + (size_t)(row0 + r) * F + (ti + 1) * TILE_COLS + ci);
            }
        }

        const float* __restrict__ tp = &ltest[ti * TILE_COLS];
        const float* __restrict__ rp = &ltile[bb][tileRow * LSTRIDE];

        #pragma unroll
        for (int kk = 0; kk < TILE_COLS / 4; ++kk) {     // 16 WMMAs per tile
            const int col = kk * 4 + koff;
            float a0 = rp[col], a1 = rp[col + 1];        // ds_load (8B, conflict-free)
            float t0 = tp[col], t1 = tp[col + 1];
            float d0 = (t0 - a0) + EPS;                  // reference: test - train + eps
            float d1 = (t1 - a1) + EPS;
            v2f av = {d0 * d0, d1 * d1};
            // D[i][j] += sum_k d2[i][k] * 1  -> row sums replicated over N
            c = __builtin_amdgcn_wmma_f32_16x16x4_f32(
                    false, av, false, ones, (short)0, c, false, false);
        }

        async_wait_all();    // next tile resident; all waves done reading bb
        __syncthreads();
    }

    // C/D layout: VGPR v, lanes 0-15 -> M=v ; lanes 16-31 -> M=v+8 (N = lane&15).
    // Lane 0 holds rows 0..7 (col 0), lane 16 holds rows 8..15 -> no shuffles.
    const int rowBase = row0 + wave * 16;
    if (lane == 0) {
        #pragma unroll
        for (int q = 0; q < 8; ++q)
            diag[rowBase + q] = expf(-sqrtf(c[q]));
    } else if (lane == 16) {
        #pragma unroll
        for (int q = 0; q < 8; ++q)
            diag[rowBase + 8 + q] = expf(-sqrtf(c[q]));
    }
}

// ---------------------------------------------------------------------------
// Kernel 2: lower-median select (bitonic sort of 16384 floats in LDS, one
// workgroup), then loss = sum_{d>=med} d*r^2 / sum_{d>=med} d.
// ---------------------------------------------------------------------------
__global__ __launch_bounds__(1024) void median_loss_kernel(
    const float* __restrict__ diag, const float* __restrict__ pred,
    const float* __restrict__ targ, float* __restrict__ out, int N) {

    extern __shared__ float s[];                        // N floats (64 KB)
    const int tid = threadIdx.x;
    const int nt  = blockDim.x;

    for (int i = tid; i < N; i += nt) s[i] = diag[i];
    __syncthreads();

    // bitonic sort, ascending
    for (int k = 2; k <= N; k <<= 1) {
        for (int j = k >> 1; j > 0; j >>= 1) {
            for (int i = tid; i < N; i += nt) {
                int ixj = i ^ j;
                if (ixj > i) {
                    float a = s[i], b = s[ixj];
                    bool up = ((i & k) == 0);
                    if ((a > b) == up) { s[i] = b; s[ixj] = a; }
                }
            }
            __syncthreads();
        }
    }

    const float med = s[(N - 1) >> 1];                  // lower median
    __syncthreads();

    float num = 0.0f, den = 0.0f;
    for (int i = tid; i < N; i += nt) {
        float dg = diag[i];
        float keep = (dg < med) ? 0.0f : dg;            // reference thresholding
        float r = targ[i] - pred[i];
        num += keep * r * r;
        den += keep;
    }

    // tree reduction (reuse sort LDS)
    s[tid] = num;
    s[nt + tid] = den;
    __syncthreads();
    for (int off = nt >> 1; off > 0; off >>= 1) {
        if (tid < off) {
            s[tid]      += s[tid + off];
            s[nt + tid] += s[nt + tid + off];
        }
        __syncthreads();
    }
    if (tid == 0) out[0] = s[0] / s[nt];
}

// ---------------------------------------------------------------------------
extern "C" void kernel_launch(void* const* d_in, const int* in_sizes, int n_in,
                              void* d_out, int out_size, void* d_ws, size_t ws_size,
                              hipStream_t stream) {
    const float* pred  = (const float*)d_in[0];
    const float* targ  = (const float*)d_in[1];
    const float* train = (const float*)d_in[2];
    const float* test  = (const float*)d_in[3];
    const int B = in_sizes[0];      // 16384
    const int F = in_sizes[3];      // 4096

    float* diag = (float*)d_ws;     // B floats of scratch

    dist_diag_kernel<<<dim3(B / ROWS_PER_WG), dim3(128), 0, stream>>>(
        train, test, diag, F);

    median_loss_kernel<<<dim3(1), dim3(1024), (size_t)B * sizeof(float), stream>>>(
        diag, pred, targ, (float*)d_out, B);
}